// DQN_11312943857936
// MI455X (gfx1250) — compile-verified
//
#include <hip/hip_runtime.h>
#include <hip/hip_bf16.h>

#define NODES  50000
#define FEAT   128
#define HID    128
#define EDGES  1600000
#define GRAPHS 64
#define ACT    8

typedef float v2f __attribute__((ext_vector_type(2)));
typedef float v8f __attribute__((ext_vector_type(8)));

// -------- init: deg = 1 (self loop), zero int histogram/cursor + pool accums --
__global__ void k0_init(float* __restrict__ deg, int* __restrict__ indeg,
                        int* __restrict__ cursor, float* __restrict__ psum,
                        float* __restrict__ cnt) {
  int i = blockIdx.x * 256 + threadIdx.x;
  if (i < NODES) { deg[i] = 1.0f; indeg[i] = 0; cursor[i] = 0; }
  if (i < GRAPHS * HID) psum[i] = 0.0f;
  if (i < GRAPHS) cnt[i] = 0.0f;
}

// -------- per-edge MLP weight + weighted degree + integer in-degree ----------
__global__ void k1_edge(const int* __restrict__ ei, const float* __restrict__ attr,
                        const float* __restrict__ We1, const float* __restrict__ be1,
                        const float* __restrict__ We2, const float* __restrict__ be2,
                        float* __restrict__ w_edge, float* __restrict__ deg,
                        int* __restrict__ indeg) {
  int e = blockIdx.x * 256 + threadIdx.x;
  if (e >= EDGES) return;
  int src = ei[e];
  int dst = ei[EDGES + e];
  float h = (float)src * We1[0] + (float)dst * We1[1] + attr[e] * We1[2] + be1[0];
  h = fmaxf(h, 0.0f);
  float s = h * We2[0] + be2[0];
  float w = 1.0f / (1.0f + __expf(-s));
  w_edge[e] = w;
  unsafeAtomicAdd(&deg[dst], w);
  atomicAdd(&indeg[dst], 1);
}

// -------- xw = x @ W_gcn via V_WMMA_F32_16X16X4_F32, A tile staged in LDS ----
// block = 8 waves; all waves share one 16x128 A tile (contiguous 8KB of x).
__global__ __launch_bounds__(256) void k2_gemm_xw(const float* __restrict__ x,
                                                  const float* __restrict__ Wg,
                                                  float* __restrict__ xw) {
  __shared__ float As[16 * FEAT];                 // 8 KB of the WGP's 320 KB LDS
  const int tid = threadIdx.x;
  const int m0 = blockIdx.x * 16;
#pragma unroll
  for (int i = 0; i < 8; ++i)                     // coalesced contiguous copy
    As[tid + i * 256] = x[(size_t)m0 * FEAT + tid + i * 256];
  __syncthreads();

  const int lane = tid & 31;
  const int wave = tid >> 5;
  const int n0 = wave * 16;
  const int ar = (lane & 15) * FEAT;              // A: row within LDS tile
  const int kh = (lane >> 4) * 2;                 // K offset 0 (lo lanes) / 2 (hi)
  const int nc = n0 + (lane & 15);                // B/C/D column per lane
  v8f acc = {0.f, 0.f, 0.f, 0.f, 0.f, 0.f, 0.f, 0.f};
  for (int k0 = 0; k0 < FEAT; k0 += 4) {
    const int ka = k0 + kh;
    v2f a = { As[ar + ka],           As[ar + ka + 1] };       // ds_load
    v2f b = { Wg[ka * HID + nc],     Wg[(ka + 1) * HID + nc] };
    acc = __builtin_amdgcn_wmma_f32_16x16x4_f32(false, a, false, b,
                                                (short)0, acc, false, false);
  }
#pragma unroll
  for (int v = 0; v < 8; ++v) {
    int row = m0 + ((lane < 16) ? v : v + 8);
    xw[row * HID + nc] = acc[v];
  }
}

// -------- single-block exclusive scan of in-degrees -> CSR row offsets -------
__global__ __launch_bounds__(1024) void k_scan(const int* __restrict__ indeg,
                                               int* __restrict__ off) {
  __shared__ int part[1024];
  const int tid = threadIdx.x;
  const int CH = (NODES + 1023) / 1024;           // 49 nodes per thread
  const int base = tid * CH;
  int local = 0;
  for (int i = 0; i < CH; ++i) {
    int idx = base + i;
    if (idx < NODES) local += indeg[idx];
  }
  part[tid] = local;
  __syncthreads();
  for (int d = 1; d < 1024; d <<= 1) {            // Hillis-Steele inclusive scan
    int v = (tid >= d) ? part[tid - d] : 0;
    __syncthreads();
    part[tid] += v;
    __syncthreads();
  }
  int run = (tid == 0) ? 0 : part[tid - 1];
  for (int i = 0; i < CH; ++i) {
    int idx = base + i;
    if (idx < NODES) { off[idx] = run; run += indeg[idx]; }
  }
  if (tid == 1023) off[NODES] = part[1023];       // == EDGES
}

// -------- dis = rsqrt(deg) ---------------------------------------------------
__global__ void k3_dis(const float* __restrict__ deg, float* __restrict__ dis) {
  int n = blockIdx.x * 256 + threadIdx.x;
  if (n < NODES) dis[n] = rsqrtf(deg[n]);         // deg >= 1 always
}

// -------- per-edge coefficient + CSR slot fill (dest-sorted edge ids) --------
__global__ void k_fill(const int* __restrict__ ei, const float* __restrict__ w_edge,
                       const float* __restrict__ dis, const int* __restrict__ off,
                       int* __restrict__ cursor, int* __restrict__ eidx,
                       float* __restrict__ coef) {
  int e = blockIdx.x * 256 + threadIdx.x;
  if (e >= EDGES) return;
  int row = ei[e];
  int col = ei[EDGES + e];
  coef[e] = dis[row] * w_edge[e] * dis[col];
  int pos = atomicAdd(&cursor[col], 1);
  eidx[off[col] + pos] = e;
}

// -------- gather: conv row in registers -> relu -> pool (no f32 scatter!) ----
__global__ __launch_bounds__(128) void k_gather(const int* __restrict__ ei,
                                                const int* __restrict__ off,
                                                const int* __restrict__ eidx,
                                                const float* __restrict__ coef,
                                                const float* __restrict__ xw,
                                                const float* __restrict__ deg,
                                                const float* __restrict__ bg,
                                                const int* __restrict__ batch,
                                                float* __restrict__ psum,
                                                float* __restrict__ cnt) {
  const int n = blockIdx.x;
  const int h = threadIdx.x;
  float acc = bg[h] + xw[n * HID + h] / deg[n];   // bias + self-loop term
  const int pend = off[n + 1];
  for (int p = off[n]; p < pend; ++p) {
    int e = eidx[p];
    int row = ei[e];                              // uniform across wave (cached)
    acc += coef[e] * xw[row * HID + h];           // coalesced 512B row gather
  }
  float v = fmaxf(acc, 0.0f);                     // relu(conv)
  int g = batch[n];
  unsafeAtomicAdd(&psum[g * HID + h], v);         // only 6.4M pooling atomics
  if (h == 0) unsafeAtomicAdd(&cnt[g], 1.0f);
}

__global__ void k6_mean(const float* __restrict__ psum, const float* __restrict__ cnt,
                        float* __restrict__ pooled) {
  int i = blockIdx.x * 256 + threadIdx.x;
  if (i >= GRAPHS * HID) return;
  int g = i / HID;
  pooled[i] = psum[i] / fmaxf(cnt[g], 1.0f);
}

// -------- h2 = relu(pooled @ W2 + b2) via V_WMMA_F32_16X16X4_F32 -------------
__global__ __launch_bounds__(256) void k7_gemm_h2(const float* __restrict__ pooled,
                                                  const float* __restrict__ W2,
                                                  const float* __restrict__ b2,
                                                  float* __restrict__ h2) {
  const int lane = threadIdx.x & 31;
  const int wave = threadIdx.x >> 5;
  const int m0 = blockIdx.x * 16;                 // 4 blocks cover M = 64
  const int n0 = wave * 16;                       // 8 waves cover N = 128
  const int mr = m0 + (lane & 15);
  const int kh = (lane >> 4) * 2;
  const int nc = n0 + (lane & 15);
  v8f acc = {0.f, 0.f, 0.f, 0.f, 0.f, 0.f, 0.f, 0.f};
  for (int k0 = 0; k0 < HID; k0 += 4) {
    const int ka = k0 + kh;
    v2f a = { pooled[mr * HID + ka], pooled[mr * HID + ka + 1] };
    v2f b = { W2[ka * HID + nc],     W2[(ka + 1) * HID + nc] };
    acc = __builtin_amdgcn_wmma_f32_16x16x4_f32(false, a, false, b,
                                                (short)0, acc, false, false);
  }
  float bias = b2[nc];
#pragma unroll
  for (int v = 0; v < 8; ++v) {
    int row = m0 + ((lane < 16) ? v : v + 8);
    h2[row * HID + nc] = fmaxf(acc[v] + bias, 0.0f);
  }
}

// -------- out = h2 @ W3 + b3  (64x8, tiny) -----------------------------------
__global__ void k8_out(const float* __restrict__ h2, const float* __restrict__ W3,
                       const float* __restrict__ b3, float* __restrict__ out) {
  int i = blockIdx.x * 256 + threadIdx.x;
  if (i >= GRAPHS * ACT) return;
  int g = i / ACT, a = i % ACT;
  float s = b3[a];
#pragma unroll 4
  for (int h = 0; h < HID; ++h) s += h2[g * HID + h] * W3[h * ACT + a];
  out[i] = s;
}

extern "C" void kernel_launch(void* const* d_in, const int* in_sizes, int n_in,
                              void* d_out, int out_size, void* d_ws, size_t ws_size,
                              hipStream_t stream) {
  const float* x    = (const float*)d_in[0];
  const float* attr = (const float*)d_in[1];
  const float* We1  = (const float*)d_in[2];
  const float* be1  = (const float*)d_in[3];
  const float* We2  = (const float*)d_in[4];
  const float* be2  = (const float*)d_in[5];
  const float* Wg   = (const float*)d_in[6];
  const float* bg   = (const float*)d_in[7];
  const float* W2   = (const float*)d_in[8];
  const float* b2   = (const float*)d_in[9];
  const float* W3   = (const float*)d_in[10];
  const float* b3   = (const float*)d_in[11];
  const int*   ei   = (const int*)d_in[12];   // (2, E): row 0 = src, row 1 = dst
  const int*   bat  = (const int*)d_in[13];
  float* out = (float*)d_out;

  // workspace carve-out (4-byte units), ~46 MB total
  float* ws     = (float*)d_ws;
  float* w_edge = ws;                               // E
  float* coef   = w_edge + EDGES;                   // E
  float* deg    = coef + EDGES;                     // N
  float* dis    = deg + NODES;                      // N
  float* xw     = dis + NODES;                      // N*H
  float* psum   = xw + (size_t)NODES * HID;         // G*H
  float* cnt    = psum + GRAPHS * HID;              // G
  float* pooled = cnt + GRAPHS;                     // G*H
  float* h2     = pooled + GRAPHS * HID;            // G*H
  int*   indeg  = (int*)(h2 + GRAPHS * HID);        // N
  int*   cursor = indeg + NODES;                    // N
  int*   off    = cursor + NODES;                   // N+1
  int*   eidx   = off + NODES + 1;                  // E

  k0_init<<<(NODES + 255) / 256, 256, 0, stream>>>(deg, indeg, cursor, psum, cnt);
  k1_edge<<<(EDGES + 255) / 256, 256, 0, stream>>>(ei, attr, We1, be1, We2, be2,
                                                   w_edge, deg, indeg);
  k2_gemm_xw<<<NODES / 16, 256, 0, stream>>>(x, Wg, xw);
  k_scan<<<1, 1024, 0, stream>>>(indeg, off);
  k3_dis<<<(NODES + 255) / 256, 256, 0, stream>>>(deg, dis);
  k_fill<<<(EDGES + 255) / 256, 256, 0, stream>>>(ei, w_edge, dis, off, cursor,
                                                  eidx, coef);
  k_gather<<<NODES, 128, 0, stream>>>(ei, off, eidx, coef, xw, deg, bg, bat,
                                      psum, cnt);
  k6_mean<<<(GRAPHS * HID + 255) / 256, 256, 0, stream>>>(psum, cnt, pooled);
  k7_gemm_h2<<<GRAPHS / 16, 256, 0, stream>>>(pooled, W2, b2, h2);
  k8_out<<<(GRAPHS * ACT + 255) / 256, 256, 0, stream>>>(h2, W3, b3, out);
}